// GCN_CCSA_24051816857602
// MI455X (gfx1250) — compile-verified
//
#include <hip/hip_runtime.h>
#include <math.h>

// GCN inference: N nodes, E edges, D=64, L=3 layers, G=128 graphs, C=10.
// Outputs: pred[G*C] then feature[G*D], concatenated in d_out.

typedef __attribute__((ext_vector_type(2))) float v2f;
typedef __attribute__((ext_vector_type(8))) float v8f;

#ifndef __has_builtin
#define __has_builtin(x) 0
#endif
#if __has_builtin(__builtin_amdgcn_wmma_f32_16x16x4_f32)
#define GCN_HAVE_WMMA_F32X4 1
#else
#define GCN_HAVE_WMMA_F32X4 0
#endif

#define D_FEAT 64
#define G_GRAPHS 128
#define C_CLASSES 10
#define EPS_BN 1e-5f

// ---------------- utility kernels ----------------

__global__ __launch_bounds__(256) void k_fill1(float* __restrict__ p, int n) {
  int i = blockIdx.x * 256 + threadIdx.x;
  if (i < n) p[i] = 1.0f;
}

__global__ __launch_bounds__(256) void k_zero(float* __restrict__ p, int n) {
  int i = blockIdx.x * 256 + threadIdx.x;
  if (i < n) p[i] = 0.0f;
}

__global__ __launch_bounds__(256) void k_deg(const int* __restrict__ dst,
                                             float* __restrict__ deg, int ne) {
  int e = blockIdx.x * 256 + threadIdx.x;
  if (e < ne) atomicAdd(&deg[dst[e]], 1.0f);
}

__global__ __launch_bounds__(256) void k_rsqrt_inplace(float* __restrict__ p, int n) {
  int i = blockIdx.x * 256 + threadIdx.x;
  if (i < n) p[i] = rsqrtf(p[i]);
}

// ---------------- GEMM: Y[N,64] = X[N,64] @ W[64,64] ----------------
// One wave computes a 16-row tile of Y (all 64 columns) via V_WMMA_F32_16X16X4_F32.
// Lane layouts per CDNA5 ISA 7.12.2:
//   A 16x4 : lanes 0-15 -> M=lane, VGPR{0,1}=K{0,1}; lanes 16-31 -> K{2,3}
//   B 4x16 : lanes 0-15 -> N=lane, VGPR{0,1}=K{0,1}; lanes 16-31 -> K{2,3}
//   C 16x16: VGPR v -> M = v + 8*(lane/16), N = lane%16
#if GCN_HAVE_WMMA_F32X4
__global__ __launch_bounds__(256) void k_gemm64_wmma(const float* __restrict__ X,
                                                     const float* __restrict__ W,
                                                     float* __restrict__ Y,
                                                     int nTiles) {
  int wave = threadIdx.x >> 5;
  int lane = threadIdx.x & 31;
  int tile = blockIdx.x * 8 + wave;        // 16-row tile index (wave-uniform guard)
  if (tile >= nTiles) return;

  int m15   = lane & 15;
  int khalf = lane >> 4;                   // 0 or 1
  const float* xrow = X + (size_t)(tile * 16 + m15) * D_FEAT;

  v8f acc0 = {}, acc1 = {}, acc2 = {}, acc3 = {};
#pragma unroll
  for (int k = 0; k < D_FEAT; k += 4) {
    v2f a;
    a.x = xrow[k + 2 * khalf + 0];
    a.y = xrow[k + 2 * khalf + 1];
    const float* wp = W + (size_t)(k + 2 * khalf) * D_FEAT + m15;
    v2f b0, b1, b2, b3;
    b0.x = wp[0];  b0.y = wp[D_FEAT + 0];
    b1.x = wp[16]; b1.y = wp[D_FEAT + 16];
    b2.x = wp[32]; b2.y = wp[D_FEAT + 32];
    b3.x = wp[48]; b3.y = wp[D_FEAT + 48];
    acc0 = __builtin_amdgcn_wmma_f32_16x16x4_f32(false, a, false, b0, (short)0, acc0, false, false);
    acc1 = __builtin_amdgcn_wmma_f32_16x16x4_f32(false, a, false, b1, (short)0, acc1, false, false);
    acc2 = __builtin_amdgcn_wmma_f32_16x16x4_f32(false, a, false, b2, (short)0, acc2, false, false);
    acc3 = __builtin_amdgcn_wmma_f32_16x16x4_f32(false, a, false, b3, (short)0, acc3, false, false);
  }
  // store: row = tile*16 + 8*khalf + v, col = ntile*16 + m15
  float* yb = Y + (size_t)(tile * 16 + 8 * khalf) * D_FEAT;
#pragma unroll
  for (int v = 0; v < 8; ++v) {
    yb[(size_t)v * D_FEAT + 0 + m15]  = acc0[v];
    yb[(size_t)v * D_FEAT + 16 + m15] = acc1[v];
    yb[(size_t)v * D_FEAT + 32 + m15] = acc2[v];
    yb[(size_t)v * D_FEAT + 48 + m15] = acc3[v];
  }
}
#else
__global__ __launch_bounds__(256) void k_gemm64_fallback(const float* __restrict__ X,
                                                         const float* __restrict__ W,
                                                         float* __restrict__ Y,
                                                         int nElems) {
  int i = blockIdx.x * 256 + threadIdx.x;
  if (i >= nElems) return;
  int row = i >> 6, col = i & 63;
  const float* xr = X + (size_t)row * D_FEAT;
  float acc = 0.0f;
#pragma unroll
  for (int k = 0; k < D_FEAT; ++k) acc = fmaf(xr[k], W[k * D_FEAT + col], acc);
  Y[i] = acc;
}
#endif

// ---------------- self-loop init: h[n,:] = tmp[n,:] * dinv[n]^2 ----------------
__global__ __launch_bounds__(256) void k_selfloop(const float* __restrict__ tmp,
                                                  const float* __restrict__ dinv,
                                                  float* __restrict__ h, int nElems) {
  int i = blockIdx.x * 256 + threadIdx.x;
  if (i >= nElems) return;
  float di = dinv[i >> 6];
  h[i] = tmp[i] * di * di;
}

// ---------------- edge scatter: one wave per edge, 2 floats per lane -----------
__global__ __launch_bounds__(256) void k_edge_scatter(const int* __restrict__ src,
                                                      const int* __restrict__ dst,
                                                      const float* __restrict__ dinv,
                                                      const float* __restrict__ msg,
                                                      float* __restrict__ out, int ne) {
  int e = blockIdx.x * 8 + (threadIdx.x >> 5);
  if (e >= ne) return;
  int lane = threadIdx.x & 31;
  int s = src[e], d = dst[e];
  float norm = dinv[s] * dinv[d];
  const float* mp = msg + (size_t)s * D_FEAT + lane * 2;
  float* op = out + (size_t)d * D_FEAT + lane * 2;
  atomicAdd(op + 0, mp[0] * norm);
  atomicAdd(op + 1, mp[1] * norm);
}

// ---------------- bias + BatchNorm(inference) + ReLU, in place -----------------
__global__ __launch_bounds__(256) void k_bias_bn_relu(float* __restrict__ h,
                                                      const float* __restrict__ b,
                                                      const float* __restrict__ gamma,
                                                      const float* __restrict__ beta,
                                                      const float* __restrict__ mean,
                                                      const float* __restrict__ var,
                                                      int nElems) {
  int i = blockIdx.x * 256 + threadIdx.x;
  if (i >= nElems) return;
  int d = i & 63;
  float scale = gamma[d] * rsqrtf(var[d] + EPS_BN);
  float v = (h[i] + b[d] - mean[d]) * scale + beta[d];
  h[i] = v > 0.0f ? v : 0.0f;
}

// ---------------- global mean pool (atomic accumulate) -------------------------
__global__ __launch_bounds__(256) void k_pool(const float* __restrict__ h,
                                              const int* __restrict__ batch,
                                              float* __restrict__ sums,
                                              float* __restrict__ counts, int nNodes) {
  int node = blockIdx.x * 8 + (threadIdx.x >> 5);
  if (node >= nNodes) return;
  int lane = threadIdx.x & 31;
  int g = batch[node];
  const float* hp = h + (size_t)node * D_FEAT + lane * 2;
  float* sp = sums + (size_t)g * D_FEAT + lane * 2;
  atomicAdd(sp + 0, hp[0]);
  atomicAdd(sp + 1, hp[1]);
  if (lane == 0) atomicAdd(&counts[g], 1.0f);
}

// ---------------- finalize: mean, L2-normalize, classifier ---------------------
__global__ __launch_bounds__(128) void k_finalize(const float* __restrict__ sums,
                                                  const float* __restrict__ counts,
                                                  const float* __restrict__ clfW,
                                                  const float* __restrict__ clfb,
                                                  float* __restrict__ pred,
                                                  float* __restrict__ feat) {
  int g = threadIdx.x;
  if (g >= G_GRAPHS) return;
  float inv = 1.0f / fmaxf(counts[g], 1.0f);
  float f[D_FEAT];
  float ss = 0.0f;
#pragma unroll
  for (int d = 0; d < D_FEAT; ++d) {
    float v = sums[g * D_FEAT + d] * inv;
    f[d] = v;
    ss += v * v;
  }
  float innrm = 1.0f / fmaxf(sqrtf(ss), 1e-12f);
#pragma unroll
  for (int d = 0; d < D_FEAT; ++d) {
    f[d] *= innrm;
    feat[g * D_FEAT + d] = f[d];
  }
#pragma unroll
  for (int c = 0; c < C_CLASSES; ++c) {
    float acc = clfb[c];
#pragma unroll
    for (int d = 0; d < D_FEAT; ++d) acc = fmaf(f[d], clfW[d * C_CLASSES + c], acc);
    pred[g * C_CLASSES + c] = acc;
  }
}

// ---------------- launch ----------------

static inline size_t align256(size_t x) { return (x + 255) & ~(size_t)255; }

extern "C" void kernel_launch(void* const* d_in, const int* in_sizes, int n_in,
                              void* d_out, int out_size, void* d_ws, size_t ws_size,
                              hipStream_t stream) {
  const float* x        = (const float*)d_in[0];
  const int*   edge     = (const int*)d_in[1];   // [2, E] row-major
  const int*   batch    = (const int*)d_in[2];
  const float* Ws       = (const float*)d_in[3]; // [L, 64, 64]
  const float* bs       = (const float*)d_in[4]; // [L, 64]
  const float* gammas   = (const float*)d_in[5];
  const float* betas    = (const float*)d_in[6];
  const float* bn_means = (const float*)d_in[7];
  const float* bn_vars  = (const float*)d_in[8];
  const float* clfW     = (const float*)d_in[9]; // [64, 10]
  const float* clfb     = (const float*)d_in[10];

  const int N = in_sizes[0] / D_FEAT;      // 100000
  const int E = in_sizes[1] / 2;           // 1200000
  const int L = in_sizes[3] / (D_FEAT * D_FEAT); // 3
  const int NE = N * D_FEAT;

  const int* srcIdx = edge;
  const int* dstIdx = edge + E;

  float* pred = (float*)d_out;                   // [G, C]
  float* feat = (float*)d_out + G_GRAPHS * C_CLASSES; // [G, D]

  // workspace carve-up
  char* ws = (char*)d_ws;
  size_t off = 0;
  float* tmp  = (float*)(ws + off); off += align256((size_t)NE * sizeof(float));
  float* h    = (float*)(ws + off); off += align256((size_t)NE * sizeof(float));
  float* dinv = (float*)(ws + off); off += align256((size_t)N * sizeof(float));
  float* sums = (float*)(ws + off); // [G*D] then counts [G], contiguous
  float* counts = sums + G_GRAPHS * D_FEAT;

  const int tpb = 256;
  int nBlk  = (N + tpb - 1) / tpb;
  int eBlk  = (E + tpb - 1) / tpb;
  int neBlk = (NE + tpb - 1) / tpb;

  // degrees with self-loops: deg = 1 + count(dst == n); dinv = rsqrt(deg)
  k_fill1<<<nBlk, tpb, 0, stream>>>(dinv, N);
  k_deg<<<eBlk, tpb, 0, stream>>>(dstIdx, dinv, E);
  k_rsqrt_inplace<<<nBlk, tpb, 0, stream>>>(dinv, N);

  const float* cur = x;
  for (int l = 0; l < L; ++l) {
#if GCN_HAVE_WMMA_F32X4
    int nTiles = N / 16;                      // N divisible by 16 (100000/16 = 6250)
    k_gemm64_wmma<<<(nTiles + 7) / 8, tpb, 0, stream>>>(cur, Ws + l * D_FEAT * D_FEAT, tmp, nTiles);
#else
    k_gemm64_fallback<<<neBlk, tpb, 0, stream>>>(cur, Ws + l * D_FEAT * D_FEAT, tmp, NE);
#endif
    k_selfloop<<<neBlk, tpb, 0, stream>>>(tmp, dinv, h, NE);
    k_edge_scatter<<<(E + 7) / 8, tpb, 0, stream>>>(srcIdx, dstIdx, dinv, tmp, h, E);
    k_bias_bn_relu<<<neBlk, tpb, 0, stream>>>(h, bs + l * D_FEAT, gammas + l * D_FEAT,
                                              betas + l * D_FEAT, bn_means + l * D_FEAT,
                                              bn_vars + l * D_FEAT, NE);
    cur = h;
  }

  int poolZero = G_GRAPHS * D_FEAT + G_GRAPHS;
  k_zero<<<(poolZero + tpb - 1) / tpb, tpb, 0, stream>>>(sums, poolZero);
  k_pool<<<(N + 7) / 8, tpb, 0, stream>>>(h, batch, sums, counts, N);
  k_finalize<<<1, 128, 0, stream>>>(sums, counts, clfW, clfb, pred, feat);
}